// FocusCalLoss_86947317941201
// MI455X (gfx1250) — compile-verified
//
#include <hip/hip_runtime.h>

#define N_BINS 15
#define TILE   4096
#define BLOCK  256
#define NWAVE  (BLOCK / 32)

typedef unsigned int u32x4 __attribute__((ext_vector_type(4)));
typedef int          i32x8 __attribute__((ext_vector_type(8)));
typedef int          i32x4 __attribute__((ext_vector_type(4)));

#if defined(__has_builtin)
#if __has_builtin(__builtin_amdgcn_tensor_load_to_lds)
#define HAVE_TDM 1
#endif
#if __has_builtin(__builtin_amdgcn_s_wait_tensorcnt)
#define HAVE_TENSORCNT_WAIT 1
#endif
#endif
#ifndef HAVE_TDM
#define HAVE_TDM 0
#endif
#ifndef HAVE_TENSORCNT_WAIT
#define HAVE_TENSORCNT_WAIT 0
#endif

#if HAVE_TDM
// 1-D TDM copy: nelem 4-byte elements from global -> LDS.
// D# layout per CDNA5 ISA (group0 128b, group1 256b).
// Toolchain here is the 6-arg (clang-23 / therock-10.0) form:
//   (u32x4 g0, i32x8 g1, i32x4 g2, i32x4 g3, i32x8 pad, i32 cpol)
__device__ __forceinline__ void tdm_load_1d(unsigned lds_off, const void* gptr, int nelem) {
  unsigned long long ga = (unsigned long long)gptr;
  u32x4 g0;
  g0[0] = 1u;                                              // count=1, user desc, no gather
  g0[1] = lds_off;                                         // lds_addr (bytes)
  g0[2] = (unsigned)(ga & 0xffffffffull);                  // global_addr[31:0]
  g0[3] = (unsigned)((ga >> 32) & 0x01ffffffull)           // global_addr[56:32]
        | (2u << 30);                                      // type=2 ("image")
  i32x8 g1;
  g1[0] = 0x00020000;                                      // workgroup_mask=0, data_size=2 (4B)
  g1[1] = (int)(((unsigned)nelem & 0xffffu) << 16);        // tensor_dim0[15:0] @ bits 63:48
  g1[2] = (int)((((unsigned)nelem >> 16) & 0xffffu)        // tensor_dim0[31:16] @ bits 79:64
        | (1u << 16));                                     // tensor_dim1 = 1
  g1[3] = (int)(((unsigned)nelem & 0xffffu) << 16);        // tile_dim0 @ bits 127:112
  g1[4] = 0;                                               // tile_dim1=0 (unused), tile_dim2=0
  g1[5] = nelem;                                           // tensor_dim0_stride[31:0]
  g1[6] = 0;                                               // stride hi / dim1_stride lo
  g1[7] = 0;
  i32x4 z4 = (i32x4)0;                                     // groups 2/3 unused (<=2D tensor)
  i32x8 z8 = (i32x8)0;
  __builtin_amdgcn_tensor_load_to_lds(g0, g1, z4, z4, z8, 0);
}
#endif

__global__ void fc_init(double* __restrict__ ws) {
  if (threadIdx.x < N_BINS + 2) ws[threadIdx.x] = 0.0;
}

__global__ void __launch_bounds__(BLOCK) fc_main(const float* __restrict__ p_hat,
                                                 const float* __restrict__ u_hat,
                                                 const int*   __restrict__ targets,
                                                 double* __restrict__ ws, int n) {
  __shared__ float s_p[TILE];
  __shared__ float s_u[TILE];
  __shared__ int   s_t[TILE];
  __shared__ float s_bins[NWAVE][16];   // per-wave replicated ECE bins (15 used, pad 16)
  __shared__ float s_red[NWAVE][2];

  const int tid  = threadIdx.x;
  const int wave = tid >> 5;
  const int lane = tid & 31;
  const long long base = (long long)blockIdx.x * TILE;
  int rem = n - (int)base;
  if (rem > TILE) rem = TILE;
  if (rem <= 0) return;

  if (tid < NWAVE * 16) ((float*)s_bins)[tid] = 0.0f;

#if HAVE_TDM
  if (wave == 0) {
    tdm_load_1d((unsigned)(unsigned long long)s_p, p_hat + base, rem);
    tdm_load_1d((unsigned)(unsigned long long)s_u, u_hat + base, rem);
    tdm_load_1d((unsigned)(unsigned long long)s_t, targets + base, rem);
#if HAVE_TENSORCNT_WAIT
    __builtin_amdgcn_s_wait_tensorcnt(0);
#else
    asm volatile("s_wait_tensorcnt 0x0" ::: "memory");
#endif
  }
  __syncthreads();
#else
  for (int k = tid; k < rem; k += BLOCK) {
    s_p[k] = p_hat[base + k];
    s_u[k] = u_hat[base + k];
    s_t[k] = targets[base + k];
  }
  __syncthreads();
#endif

  float facc = 0.0f, cacc = 0.0f;
  for (int k = tid; k < rem; k += BLOCK) {
    float p  = s_p[k];
    float uu = s_u[k];
    int   tv = s_t[k];
    bool  pos = (tv == 1);

    // focal: -alpha * (1-pt)^gamma * log(pt + 1e-12), gamma = 2
    float pt = pos ? p : 1.0f - p;
    float a  = pos ? 1.0f : 2.0f;
    float om = 1.0f - pt;
    facc -= a * om * om * logf(pt + 1e-12f);

    // calibration penalty
    float pen = 0.0f;
    if (pos) { if (p < 0.7f) pen = uu * uu; }
    else     { if (p > 0.7f) { float w = 1.0f - uu; pen = w * w; } }
    cacc += pen;

    // ECE bin: b = clip(ceil(p*15)-1, 0, 14); accumulate (y - p) per bin
    int b = (int)ceilf(p * 15.0f) - 1;
    b = b < 0 ? 0 : (b > N_BINS - 1 ? N_BINS - 1 : b);
    float d = (pos ? 1.0f : 0.0f) - p;
    atomicAdd(&s_bins[wave][b], d);   // ds_add_f32
  }

  // wave32 reduction of focal / cp partials
  for (int o = 16; o > 0; o >>= 1) {
    facc += __shfl_down(facc, o, 32);
    cacc += __shfl_down(cacc, o, 32);
  }
  if (lane == 0) { s_red[wave][0] = facc; s_red[wave][1] = cacc; }
  __syncthreads();

  if (tid == 0) {
    float fs = 0.0f, cs = 0.0f;
    for (int w = 0; w < NWAVE; ++w) { fs += s_red[w][0]; cs += s_red[w][1]; }
    atomicAdd(&ws[N_BINS + 0], (double)fs);   // global_atomic_add_f64
    atomicAdd(&ws[N_BINS + 1], (double)cs);
  }
  if (tid < N_BINS) {
    float s = 0.0f;
    for (int w = 0; w < NWAVE; ++w) s += s_bins[w][tid];
    atomicAdd(&ws[tid], (double)s);
  }
}

__global__ void fc_finalize(const double* __restrict__ ws, float* __restrict__ out, int n) {
  int lane = threadIdx.x;
  double v = (lane < N_BINS) ? fabs(ws[lane]) : 0.0;
  for (int o = 16; o > 0; o >>= 1) v += __shfl_down(v, o, 32);
  if (lane == 0) {
    double inv = 1.0 / (double)n;
    out[0] = (float)((ws[N_BINS] + 6.0 * ws[N_BINS + 1] + 5.0 * v) * inv);
  }
}

extern "C" void kernel_launch(void* const* d_in, const int* in_sizes, int n_in,
                              void* d_out, int out_size, void* d_ws, size_t ws_size,
                              hipStream_t stream) {
  const float* p = (const float*)d_in[0];
  const float* u = (const float*)d_in[1];
  const int*   t = (const int*)d_in[2];
  int n = in_sizes[0];
  double* ws = (double*)d_ws;

  fc_init<<<1, 32, 0, stream>>>(ws);
  int blocks = (n + TILE - 1) / TILE;
  fc_main<<<blocks, BLOCK, 0, stream>>>(p, u, t, ws, n);
  fc_finalize<<<1, 32, 0, stream>>>(ws, (float*)d_out, n);
}